// TropicalLinear_23888608101008
// MI455X (gfx1250) — compile-verified
//
#include <hip/hip_runtime.h>
#include <math.h>

// Tropical (max-plus) linear layer: y[b,o] = max_i ( x[b,i] + W[o,i] )
//   x: [512, 1024] f32, W: [1024, 1024] f32, y: [512, 1024] f32
//
// VALU-bound (AI ~128 ops/byte vs 23.3 TB/s HBM). WMMA is unusable for the
// max-plus semiring, so the CDNA5 features exploited are:
//   * GLOBAL_LOAD_ASYNC_TO_LDS_B32 (per-lane scatter => transpose-on-copy),
//     double-buffered, synced with s_wait_asynccnt
//   * wave32-aware tiling + v_max3_num_f32 reduction folding

#define TL_BATCH 512
#define TL_IN    1024
#define TL_OUT   1024

#define BM 32      // batch rows per block
#define BN 64      // out cols per block
#define BK 32      // k-chunk
#define NTHREADS 256
#define NSTAGES (TL_IN / BK)      // 32
// async copies issued per thread per stage: 4 (x tile) + 8 (W tile)
#define ASYNC_PER_STAGE 12

#if defined(__HIP_DEVICE_COMPILE__) && defined(__gfx1250__) && \
    __has_builtin(__builtin_amdgcn_global_load_async_to_lds_b32)
#define USE_ASYNC 1
#else
#define USE_ASYNC 0
#endif

#if USE_ASYNC
// Builtin signature (from clang diagnostic): AS1 int* , AS3 int* , imm, imm
typedef __attribute__((address_space(1))) int gas_int_t;
typedef __attribute__((address_space(3))) int las_int_t;

__device__ __forceinline__ void tl_async_copy_f32(const float* g, float* l) {
  // one f32 per lane, arbitrary per-lane LDS destination (transpose-on-copy)
  __builtin_amdgcn_global_load_async_to_lds_b32(
      (gas_int_t*)(void*)g, (las_int_t*)(void*)l, 0, 0);
}

#if __has_builtin(__builtin_amdgcn_s_wait_asynccnt)
#define TL_ASYNC_WAIT(n) __builtin_amdgcn_s_wait_asynccnt(n)
#else
#define TL_ASYNC_WAIT(n) asm volatile("s_wait_asynccnt " #n ::: "memory")
#endif

#else  // !USE_ASYNC: synchronous fallback (still correct)
__device__ __forceinline__ void tl_async_copy_f32(const float* g, float* l) {
  *l = *g;
}
#define TL_ASYNC_WAIT(n)
#endif

__global__ __launch_bounds__(NTHREADS)
void TropicalLinear_kernel(const float* __restrict__ x,
                           const float* __restrict__ W,
                           float* __restrict__ y) {
  // k-major tiles: conflict-free DS reads (sX broadcast, sW 16 distinct banks)
  __shared__ float sX[2][BK][BM];   // 8 KB
  __shared__ float sW[2][BK][BN];   // 16 KB

  const int tid = threadIdx.x;
  const int tx  = tid & 15;         // 16 col-groups * TN=4 -> 64 cols
  const int ty  = tid >> 4;         // 16 row-groups * TM=2 -> 32 rows
  const int bn0 = blockIdx.x * BN;
  const int bm0 = blockIdx.y * BM;

  // copy-thread mapping: lane-consecutive k => coalesced 128B global reads
  const int kc = tid & 31;          // k within tile
  const int rc = tid >> 5;          // row base (0..7)

  float acc[2][4];
#pragma unroll
  for (int r = 0; r < 2; ++r)
#pragma unroll
    for (int j = 0; j < 4; ++j)
      acc[r][j] = -__builtin_huge_valf();

  // --- stage 0 prefetch ---
  {
    const int k0 = 0;
#pragma unroll
    for (int i = 0; i < 4; ++i) {
      const int b = rc + i * 8;
      tl_async_copy_f32(x + (size_t)(bm0 + b) * TL_IN + k0 + kc, &sX[0][kc][b]);
    }
#pragma unroll
    for (int i = 0; i < 8; ++i) {
      const int o = rc + i * 8;
      tl_async_copy_f32(W + (size_t)(bn0 + o) * TL_IN + k0 + kc, &sW[0][kc][o]);
    }
  }

  for (int s = 0; s < NSTAGES; ++s) {
    const int buf = s & 1;

    // prefetch next stage into the other buffer (its readers finished at the
    // trailing __syncthreads() of the previous iteration)
    if (s + 1 < NSTAGES) {
      const int k0 = (s + 1) * BK;
      const int nb = buf ^ 1;
#pragma unroll
      for (int i = 0; i < 4; ++i) {
        const int b = rc + i * 8;
        tl_async_copy_f32(x + (size_t)(bm0 + b) * TL_IN + k0 + kc,
                          &sX[nb][kc][b]);
      }
#pragma unroll
      for (int i = 0; i < 8; ++i) {
        const int o = rc + i * 8;
        tl_async_copy_f32(W + (size_t)(bn0 + o) * TL_IN + k0 + kc,
                          &sW[nb][kc][o]);
      }
#if USE_ASYNC
      TL_ASYNC_WAIT(ASYNC_PER_STAGE);  // stage s done; stage s+1 in flight
#endif
    } else {
#if USE_ASYNC
      TL_ASYNC_WAIT(0);                // drain final stage
#endif
    }
    __syncthreads();                   // tile filled cooperatively by 8 waves

    const float(*A)[BM]  = sX[buf];
    const float(*Bw)[BN] = sW[buf];

#pragma unroll
    for (int kk = 0; kk < BK; kk += 2) {
      const float2 a0 = *(const float2*)&A[kk][2 * ty];
      const float2 a1 = *(const float2*)&A[kk + 1][2 * ty];
      const float4 w0 = *(const float4*)&Bw[kk][4 * tx];
      const float4 w1 = *(const float4*)&Bw[kk + 1][4 * tx];
      const float av0[2] = {a0.x, a0.y};
      const float av1[2] = {a1.x, a1.y};
      const float wv0[4] = {w0.x, w0.y, w0.z, w0.w};
      const float wv1[4] = {w1.x, w1.y, w1.z, w1.w};
#pragma unroll
      for (int r = 0; r < 2; ++r)
#pragma unroll
        for (int j = 0; j < 4; ++j)
          // fmax(fmax(.,.),acc) -> v_max3_num_f32: 3 VALU per 2 k-steps
          acc[r][j] = fmaxf(acc[r][j],
                            fmaxf(av0[r] + wv0[j], av1[r] + wv1[j]));
    }
    __syncthreads();                   // before buffer is overwritten again
  }

  // epilogue: 4 consecutive cols -> one b128 store
#pragma unroll
  for (int r = 0; r < 2; ++r) {
    float4 o4;
    o4.x = acc[r][0]; o4.y = acc[r][1]; o4.z = acc[r][2]; o4.w = acc[r][3];
    *(float4*)&y[(size_t)(bm0 + 2 * ty + r) * TL_OUT + bn0 + 4 * tx] = o4;
  }
}

extern "C" void kernel_launch(void* const* d_in, const int* in_sizes, int n_in,
                              void* d_out, int out_size, void* d_ws,
                              size_t ws_size, hipStream_t stream) {
  const float* x = (const float*)d_in[0];   // [512, 1024]
  const float* W = (const float*)d_in[1];   // [1024, 1024]
  float* y = (float*)d_out;                 // [512, 1024]

  dim3 grid(TL_OUT / BN, TL_BATCH / BM);    // (16, 16) = 256 blocks
  dim3 block(NTHREADS);
  TropicalLinear_kernel<<<grid, block, 0, stream>>>(x, W, y);
}